// GeoLoss_32908039422213
// MI455X (gfx1250) — compile-verified
//
#include <hip/hip_runtime.h>
#include <stdint.h>

// ---------------- problem constants ----------------
#define N_TOTAL   16777216
#define K_TOP     6710886u          // int(0.4 * 16777216)

// ---------------- workspace layout (uint32 units) ----------------
#define OFF_H1    0                 // 4096-bin histogram, key bits [31:20]
#define OFF_H2    4096              // 4096-bin histogram, key bits [19:8]
#define OFF_H3    8192              // 256-bin histogram,  key bits [7:0]
#define OFF_S     8448              // scalars: 0=b1 1=rem1 2=prefix24 3=rem2 4=tau 5=ties_needed 6=list_cnt
#define OFF_TIES  8464              // 3 floats: tie pt / p / t sums
#define OFF_LIST  8472              // candidate list: LISTCAP entries x 4 u32 {idx, low8, x_bits, t_bits}
#define LISTCAP   4096
#define OFF_PART  (OFF_LIST + LISTCAP * 4)   // per-block partials of main pass
#define ZERO_N    OFF_LIST

#define NB_HIST   4096              // 4096 blk * 256 thr * 16 elem = 16M
#define NB_MAIN   2048              // 2048 blk * 256 thr * 4 elem * 8 iter = 16M
#define MAIN_ITERS 8

typedef float v2f __attribute__((ext_vector_type(2)));
typedef float v8f __attribute__((ext_vector_type(8)));

// monotonic key: larger key <=> larger float (total order, handles sign)
__device__ __forceinline__ uint32_t mono(float f) {
    uint32_t u = __float_as_uint(f);
    return (u & 0x80000000u) ? ~u : (u | 0x80000000u);
}

// sigmoid with hardware v_exp_f32 + v_rcp_f32 (1 ulp), avoids IEEE div sequence
__device__ __forceinline__ float sigmoidf_fast(float xv) {
    return __builtin_amdgcn_rcpf(1.0f + __expf(-xv));
}

// ---------------- P0: zero workspace head ----------------
__global__ void k_zero(uint32_t* __restrict__ ws) {
    int i = blockIdx.x * 256 + threadIdx.x;
    if (i < ZERO_N) ws[i] = 0u;
}

// ---------------- P1: histogram of key[31:20] ----------------
__global__ void __launch_bounds__(256) k_hist1(const float* __restrict__ c,
                                               uint32_t* __restrict__ ws) {
    __shared__ uint32_t lh[4096];
    for (int i = threadIdx.x; i < 4096; i += 256) lh[i] = 0u;
    __syncthreads();
    const float4* c4 = (const float4*)c;
    const size_t base = (size_t)blockIdx.x * 1024;
    #pragma unroll
    for (int j = 0; j < 4; ++j) {
        float4 v = c4[base + (size_t)j * 256 + threadIdx.x];
        atomicAdd(&lh[mono(v.x) >> 20], 1u);
        atomicAdd(&lh[mono(v.y) >> 20], 1u);
        atomicAdd(&lh[mono(v.z) >> 20], 1u);
        atomicAdd(&lh[mono(v.w) >> 20], 1u);
    }
    __syncthreads();
    uint32_t* hist = ws + OFF_H1;
    for (int i = threadIdx.x; i < 4096; i += 256)
        if (lh[i]) atomicAdd(&hist[i], lh[i]);
}

// ---------------- P2: find boundary bin at level 1 ----------------
__global__ void k_scan1(uint32_t* __restrict__ ws) {
    if (threadIdx.x != 0) return;
    const uint32_t* h = ws + OFF_H1;
    uint32_t* s = ws + OFF_S;
    uint32_t acc = 0;
    for (int b = 4095;; --b) {
        uint32_t v = h[b];
        if (acc + v >= K_TOP) { s[0] = (uint32_t)b; s[1] = K_TOP - acc; break; }
        acc += v;
    }
}

// ---------------- P3: histogram of key[19:8] within bin b1 ----------------
__global__ void __launch_bounds__(256) k_hist2(const float* __restrict__ c,
                                               uint32_t* __restrict__ ws) {
    __shared__ uint32_t lh[4096];
    for (int i = threadIdx.x; i < 4096; i += 256) lh[i] = 0u;
    __syncthreads();
    const uint32_t b1 = ws[OFF_S + 0];
    const float4* c4 = (const float4*)c;
    const size_t base = (size_t)blockIdx.x * 1024;
    #pragma unroll
    for (int j = 0; j < 4; ++j) {
        float4 v = c4[base + (size_t)j * 256 + threadIdx.x];
        float vv[4] = {v.x, v.y, v.z, v.w};
        #pragma unroll
        for (int q = 0; q < 4; ++q) {
            uint32_t key = mono(vv[q]);
            if ((key >> 20) == b1) atomicAdd(&lh[(key >> 8) & 0xFFFu], 1u);
        }
    }
    __syncthreads();
    uint32_t* hist = ws + OFF_H2;
    for (int i = threadIdx.x; i < 4096; i += 256)
        if (lh[i]) atomicAdd(&hist[i], lh[i]);
}

// ---------------- P4: find boundary sub-bin at level 2 ----------------
__global__ void k_scan2(uint32_t* __restrict__ ws) {
    if (threadIdx.x != 0) return;
    const uint32_t* h = ws + OFF_H2;
    uint32_t* s = ws + OFF_S;
    uint32_t need = s[1];
    uint32_t acc = 0;
    for (int b = 4095;; --b) {
        uint32_t v = h[b];
        if (acc + v >= need) {
            s[2] = (s[0] << 12) | (uint32_t)b;   // 24-bit prefix
            s[3] = need - acc;
            break;
        }
        acc += v;
    }
}

// ---------------- P5: collect elements matching 24-bit prefix ----------------
__global__ void __launch_bounds__(256) k_collect(const float* __restrict__ x,
                                                 const float* __restrict__ t,
                                                 const float* __restrict__ c,
                                                 uint32_t* __restrict__ ws) {
    const uint32_t prefix = ws[OFF_S + 2];
    uint32_t* hist3 = ws + OFF_H3;
    uint32_t* list  = ws + OFF_LIST;
    const float4* c4 = (const float4*)c;
    const size_t base = (size_t)blockIdx.x * 1024;
    #pragma unroll
    for (int j = 0; j < 4; ++j) {
        size_t f4i = base + (size_t)j * 256 + threadIdx.x;
        float4 v = c4[f4i];
        float vv[4] = {v.x, v.y, v.z, v.w};
        #pragma unroll
        for (int q = 0; q < 4; ++q) {
            uint32_t key = mono(vv[q]);
            if ((key >> 8) == prefix) {
                atomicAdd(&hist3[key & 0xFFu], 1u);
                uint32_t pos = atomicAdd(&ws[OFF_S + 6], 1u);
                if (pos < LISTCAP) {
                    uint32_t idx = (uint32_t)(f4i * 4 + (size_t)q);
                    uint32_t* e = &list[(size_t)pos * 4];
                    e[0] = idx;
                    e[1] = key & 0xFFu;
                    e[2] = __float_as_uint(x[idx]);
                    e[3] = __float_as_uint(t[idx]);
                }
            }
        }
    }
}

// ---------------- P6: exact threshold + index-ordered tie contributions ----------------
__global__ void k_scan3(uint32_t* __restrict__ ws) {
    if (threadIdx.x != 0) return;
    const uint32_t* h = ws + OFF_H3;
    uint32_t* s = ws + OFF_S;
    float* ties = (float*)(ws + OFF_TIES);
    uint32_t need = s[3];
    uint32_t acc = 0;
    int b = 255;
    for (;; --b) {
        uint32_t v = h[b];
        if (acc + v >= need) break;
        acc += v;
    }
    uint32_t ties_needed = need - acc;
    uint32_t tau = (s[2] << 8) | (uint32_t)b;
    s[4] = tau;
    s[5] = ties_needed;

    // gather ties (key == tau) from the candidate list
    uint32_t cnt = s[6]; if (cnt > LISTCAP) cnt = LISTCAP;
    const uint32_t* list = ws + OFF_LIST;
    uint32_t ti[256]; float tx[256]; float tt[256];
    uint32_t nt = 0;
    for (uint32_t i = 0; i < cnt; ++i) {
        const uint32_t* e = &list[(size_t)i * 4];
        if (e[1] == (uint32_t)b && nt < 256u) {
            ti[nt] = e[0];
            tx[nt] = __uint_as_float(e[2]);
            tt[nt] = __uint_as_float(e[3]);
            ++nt;
        }
    }
    // insertion sort by original index (matches top_k lowest-index tie-break)
    for (uint32_t i = 1; i < nt; ++i) {
        uint32_t ki = ti[i]; float kx = tx[i]; float kt = tt[i];
        int j = (int)i - 1;
        while (j >= 0 && ti[j] > ki) {
            ti[j + 1] = ti[j]; tx[j + 1] = tx[j]; tt[j + 1] = tt[j]; --j;
        }
        ti[j + 1] = ki; tx[j + 1] = kx; tt[j + 1] = kt;
    }
    uint32_t take = (ties_needed < nt) ? ties_needed : nt;
    float spt = 0.0f, sp = 0.0f, st = 0.0f;
    for (uint32_t i = 0; i < take; ++i) {
        float p = sigmoidf_fast(tx[i]);
        spt += p * tt[i]; sp += p; st += tt[i];
    }
    ties[0] = spt; ties[1] = sp; ties[2] = st;
}

// ---------------- P9: main masked accumulation via WMMA f32 reduction ----------------
__global__ void __launch_bounds__(256) k_main(const float* __restrict__ x,
                                              const float* __restrict__ t,
                                              const float* __restrict__ c,
                                              const uint32_t* __restrict__ ws,
                                              float* __restrict__ partials) {
    const uint32_t tau = ws[OFF_S + 4];
    const float4* x4 = (const float4*)x;
    const float4* t4 = (const float4*)t;
    const float4* c4 = (const float4*)c;

    v2f onesA = {1.0f, 1.0f};   // A = all-ones 16x4 f32
    v8f apt = {}; v8f ap = {}; v8f at = {};

    const int gtid   = blockIdx.x * 256 + threadIdx.x;
    const int stride = NB_MAIN * 256;               // float4 stride per iteration

    #pragma unroll 2
    for (int it = 0; it < MAIN_ITERS; ++it) {
        size_t i4 = (size_t)it * stride + gtid;
        float4 cv = c4[i4];
        float4 xv = x4[i4];
        float4 tv = t4[i4];

        uint32_t k0 = mono(cv.x), k1 = mono(cv.y);
        uint32_t k2 = mono(cv.z), k3 = mono(cv.w);
        float s0 = sigmoidf_fast(xv.x);
        float s1 = sigmoidf_fast(xv.y);
        float s2 = sigmoidf_fast(xv.z);
        float s3 = sigmoidf_fast(xv.w);
        float p0 = (k0 > tau) ? s0 : 0.0f;
        float p1 = (k1 > tau) ? s1 : 0.0f;
        float p2 = (k2 > tau) ? s2 : 0.0f;
        float p3 = (k3 > tau) ? s3 : 0.0f;
        float q0 = (k0 > tau) ? tv.x : 0.0f;
        float q1 = (k1 > tau) ? tv.y : 0.0f;
        float q2 = (k2 > tau) ? tv.z : 0.0f;
        float q3 = (k3 > tau) ? tv.w : 0.0f;

        v2f bp0  = {p0, p1};       v2f bp1  = {p2, p3};
        v2f bq0  = {q0, q1};       v2f bq1  = {q2, q3};
        v2f bpq0 = {p0 * q0, p1 * q1};
        v2f bpq1 = {p2 * q2, p3 * q3};

        // D = ones(16x4) x B(4x16) + C  => every B element lands in exactly
        // one column-sum; wave-wide f32 reduction of 64 elems per wmma.
        apt = __builtin_amdgcn_wmma_f32_16x16x4_f32(false, onesA, false, bpq0,
                                                    (short)0, apt, false, false);
        ap  = __builtin_amdgcn_wmma_f32_16x16x4_f32(false, onesA, false, bp0,
                                                    (short)0, ap,  false, false);
        at  = __builtin_amdgcn_wmma_f32_16x16x4_f32(false, onesA, false, bq0,
                                                    (short)0, at,  false, false);
        apt = __builtin_amdgcn_wmma_f32_16x16x4_f32(false, onesA, false, bpq1,
                                                    (short)0, apt, false, false);
        ap  = __builtin_amdgcn_wmma_f32_16x16x4_f32(false, onesA, false, bp1,
                                                    (short)0, ap,  false, false);
        at  = __builtin_amdgcn_wmma_f32_16x16x4_f32(false, onesA, false, bq1,
                                                    (short)0, at,  false, false);
    }

    // acc[0]: lanes 0-15 = column sums, lanes 16-31 = identical copy (row 8)
    float rpt = apt[0], rp = ap[0], rt = at[0];
    #pragma unroll
    for (int off = 16; off >= 1; off >>= 1) {
        rpt += __shfl_xor(rpt, off, 32);
        rp  += __shfl_xor(rp,  off, 32);
        rt  += __shfl_xor(rt,  off, 32);
    }
    __shared__ float red[8][3];
    int wid = threadIdx.x >> 5;
    if ((threadIdx.x & 31) == 0) {
        red[wid][0] = 0.5f * rpt;   // halve: both lane halves hold full column sums
        red[wid][1] = 0.5f * rp;
        red[wid][2] = 0.5f * rt;
    }
    __syncthreads();
    if (threadIdx.x == 0) {
        float a = 0.0f, b = 0.0f, d = 0.0f;
        #pragma unroll
        for (int w = 0; w < 8; ++w) { a += red[w][0]; b += red[w][1]; d += red[w][2]; }
        partials[blockIdx.x * 3 + 0] = a;
        partials[blockIdx.x * 3 + 1] = b;
        partials[blockIdx.x * 3 + 2] = d;
    }
}

// ---------------- P10: deterministic final reduction + loss ----------------
__global__ void __launch_bounds__(256) k_final(const float* __restrict__ partials,
                                               const uint32_t* __restrict__ ws,
                                               float* __restrict__ out) {
    __shared__ float s0[256], s1[256], s2[256];
    float a0 = 0.0f, a1 = 0.0f, a2 = 0.0f;
    for (int i = threadIdx.x; i < NB_MAIN; i += 256) {
        a0 += partials[3 * i + 0];
        a1 += partials[3 * i + 1];
        a2 += partials[3 * i + 2];
    }
    s0[threadIdx.x] = a0; s1[threadIdx.x] = a1; s2[threadIdx.x] = a2;
    __syncthreads();
    for (int st = 128; st > 0; st >>= 1) {
        if (threadIdx.x < st) {
            s0[threadIdx.x] += s0[threadIdx.x + st];
            s1[threadIdx.x] += s1[threadIdx.x + st];
            s2[threadIdx.x] += s2[threadIdx.x + st];
        }
        __syncthreads();
    }
    if (threadIdx.x == 0) {
        const float* ties = (const float*)(ws + OFF_TIES);
        float pt = s0[0] + ties[0];
        float p  = s1[0] + ties[1];
        float tt = s2[0] + ties[2];
        float dice = (2.0f * pt + 1.0f) / (p + tt + 1.0f);
        out[0] = 1.0f - dice;
    }
}

// ---------------- launcher ----------------
extern "C" void kernel_launch(void* const* d_in, const int* in_sizes, int n_in,
                              void* d_out, int out_size, void* d_ws, size_t ws_size,
                              hipStream_t stream) {
    (void)in_sizes; (void)n_in; (void)out_size; (void)ws_size;
    const float* x = (const float*)d_in[0];   // raw logits
    const float* t = (const float*)d_in[1];   // targets
    const float* c = (const float*)d_in[2];   // curvatures
    uint32_t* ws   = (uint32_t*)d_ws;
    float* out     = (float*)d_out;
    float* partials = (float*)(ws + OFF_PART);

    k_zero   <<<(ZERO_N + 255) / 256, 256, 0, stream>>>(ws);
    k_hist1  <<<NB_HIST, 256, 0, stream>>>(c, ws);
    k_scan1  <<<1, 32, 0, stream>>>(ws);
    k_hist2  <<<NB_HIST, 256, 0, stream>>>(c, ws);
    k_scan2  <<<1, 32, 0, stream>>>(ws);
    k_collect<<<NB_HIST, 256, 0, stream>>>(x, t, c, ws);
    k_scan3  <<<1, 32, 0, stream>>>(ws);
    k_main   <<<NB_MAIN, 256, 0, stream>>>(x, t, c, ws, partials);
    k_final  <<<1, 256, 0, stream>>>(partials, ws, out);
}